// ExtendedKalmanFilter_23837068493060
// MI455X (gfx1250) — compile-verified
//
#include <hip/hip_runtime.h>
#include <hip/hip_bf16.h>

// EKF scan: B=4096 chains, T=256 steps, M=8 state, N=4 obs.
// 2 chains per wave32; 8x8 products packed block-diagonally into 16x16
// V_WMMA_F32_16X16X4_F32 tiles (A=[F0;F1] (16xK), B=[P0|P1] (Kx16),
// diagonal D blocks valid).
//
// Layout assumptions (from CDNA5 ISA):
//  A (16x4 f32):  elem(m,k) -> lane = (m%16) + 16*(k/2), vgpr = k%2
//  B (4x16 f32):  elem(k,n) -> lane = (n%16) + 16*(k/2), vgpr = k%2  (mirror of A)
//  C/D (16x16):   elem(m,n) -> vgpr = m%8, lane = n + 16*(m/8)
// => valid D blocks: chain0 at lanes 0-7, chain1 at lanes 24-31 (vgpr=row, lane=col)
// => a symmetric-role matrix (F, H) has identical register contents as A- and B-operand.

#define TT 256
#define N4 4

typedef __attribute__((ext_vector_type(2))) float v2f;
typedef __attribute__((ext_vector_type(8))) float v8f;

__device__ __forceinline__ float shf(float v, int srcLane) {
  return __int_as_float(__builtin_amdgcn_ds_bpermute(srcLane << 2, __float_as_int(v)));
}

__device__ __forceinline__ v8f wmma4(float a0, float a1, float b0, float b1, v8f c) {
  v2f A, B;
  A[0] = a0; A[1] = a1;
  B[0] = b0; B[1] = b1;
  return __builtin_amdgcn_wmma_f32_16x16x4_f32(false, A, false, B, (short)0, c, false, false);
}

// gfx1250 hardware transcendentals (TRANS pipe, co-executes with WMMA)
__device__ __forceinline__ float fast_tanh(float x) {
#if __has_builtin(__builtin_amdgcn_tanhf)
  return __builtin_amdgcn_tanhf(x);              // v_tanh_f32
#else
  // branchless: 1 - 2/(e^{2x}+1); saturates correctly via inf/0
  const float e = __expf(2.0f * x);
  return 1.0f - 2.0f * __builtin_amdgcn_rcpf(e + 1.0f);
#endif
}
#define INV_2PI 0.15915494309189535f
__device__ __forceinline__ float fast_sin(float x) {
  return __builtin_amdgcn_sinf(x * INV_2PI);     // v_sin_f32: sin(2*pi*in)
}
__device__ __forceinline__ float fast_cos(float x) {
  return __builtin_amdgcn_cosf(x * INV_2PI);     // v_cos_f32
}

__global__ __launch_bounds__(256) void ekf1250_kernel(
    const float* __restrict__ z,  const float* __restrict__ Ag,
    const float* __restrict__ Cg, const float* __restrict__ Qg,
    const float* __restrict__ Rg, const float* __restrict__ x0g,
    const float* __restrict__ P0g, float* __restrict__ out)
{
  __shared__ float lds[8 * 640];                 // 640 floats per wave slice
  const int tid  = threadIdx.x;
  const int lane = tid & 31;
  const int wave = (blockIdx.x * blockDim.x + tid) >> 5;
  const long b0  = (long)wave * 2;               // two chains per wave
  float* W = &lds[(tid >> 5) * 640];
  // per-wave LDS regions (float offsets)
  const int RP = 0;    // P (2 x 8x8 row-major), persistent
  const int R1 = 128;  // G = F*P
  const int R2 = 256;  // P' (prior cov)
  const int R3 = 384;  // T1 = H*P'  (2 x 4x8)
  const int R4 = 448;  // PHt        (2 x 8x4)
  const int RT = 512;  // trash for junk-lane stores (128 floats)

  const int  r    = lane & 7;            // local row/col
  const int  e8   = (lane >> 3) & 1;     // chain id in A/B operand layouts
  const int  khi  = lane >> 4;           // K-half in A/B operand layouts
  const int  eD   = lane >> 4;           // chain id in D layout / K-lane layout
  const bool dval = (lane < 8) || (lane >= 24);  // valid D-layout lanes
  const bool lowh = lane < 16;
  const int  dcol = lane & 7;

  const int aBase  = e8 * 64 + r * 8 + 2 * khi;        // A-layout gather from 8x8
  const int bBase  = e8 * 64 + (2 * khi) * 8 + r;      // B-layout gather from 8x8
  const int a3Base = e8 * 32 + (r & 3) * 8 + 2 * khi;  // A-layout gather from 4x8

  // ---- wave-uniform model matrices into per-lane registers (row = lane&7) ----
  float regA[8], regC[8];
  {
    const float4 a0 = *reinterpret_cast<const float4*>(Ag + r * 8);
    const float4 a1 = *reinterpret_cast<const float4*>(Ag + r * 8 + 4);
    regA[0]=a0.x; regA[1]=a0.y; regA[2]=a0.z; regA[3]=a0.w;
    regA[4]=a1.x; regA[5]=a1.y; regA[6]=a1.z; regA[7]=a1.w;
    const float4 c0 = *reinterpret_cast<const float4*>(Cg + (r & 3) * 8);
    const float4 c1 = *reinterpret_cast<const float4*>(Cg + (r & 3) * 8 + 4);
    const float m = (r < 4) ? 1.0f : 0.0f;     // rows 4-7 of H are zero-padded
    regC[0]=c0.x*m; regC[1]=c0.y*m; regC[2]=c0.z*m; regC[3]=c0.w*m;
    regC[4]=c1.x*m; regC[5]=c1.y*m; regC[6]=c1.z*m; regC[7]=c1.w*m;
  }
  v8f Qd, Rd;
  #pragma unroll
  for (int i = 0; i < 8; ++i) Qd[i] = Qg[i * 8 + dcol];
  #pragma unroll
  for (int i = 0; i < 8; ++i) {
    float rv = Rg[(i & 3) * 4 + (dcol & 3)];
    Rd[i] = (i < 4 && dcol < 4) ? rv : 0.0f;
  }

  float xrep = x0g[r];                            // rep layout: lane (16e + i) holds x_e[i]
  for (int idx = lane; idx < 128; idx += 32) W[RP + idx] = P0g[idx & 63];

  const v8f z8 = {0,0,0,0,0,0,0,0};

  #pragma unroll 1
  for (int t = 0; t < TT; ++t) {
    // ---- nonlinear propagate: x' = tanh(Ax), u = Cx', y = sin(u) ----
    const int sb = lane & 16;                     // rep-layout broadcast base
    float xb[8];
    #pragma unroll
    for (int k = 0; k < 8; ++k) xb[k] = shf(xrep, sb + k);
    float ta = 0.0f;
    #pragma unroll
    for (int k = 0; k < 8; ++k) ta = fmaf(regA[k], xb[k], ta);
    const float xp = fast_tanh(ta);
    const float dd = 1.0f - xp * xp;              // tanh' -> F = diag(dd)*A
    float u = 0.0f;
    #pragma unroll
    for (int k = 0; k < 8; ++k) u = fmaf(regC[k], shf(xp, sb + k), u);
    const float ys = fast_sin(u);
    const float cy = fast_cos(u);                 // H = diag(cy)*C

    // ---- F and H operands (identical as A- or B-operand) ----
    const int srcAB = (e8 << 4) + r;
    const float dA  = shf(dd, srcAB);
    const float cyA = shf(cy, srcAB);
    float Fo[4], Ho[4];
    #pragma unroll
    for (int jc = 0; jc < 2; ++jc)
      #pragma unroll
      for (int v = 0; v < 2; ++v) {
        const float av = lowh ? regA[4 * jc + v] : regA[4 * jc + v + 2];
        const float cv = lowh ? regC[4 * jc + v] : regC[4 * jc + v + 2];
        Fo[2 * jc + v] = dA * av;
        Ho[2 * jc + v] = cyA * cv;
      }

    // ---- G = F*P ----
    v8f G = wmma4(Fo[0], Fo[1], W[RP + bBase + 0 ], W[RP + bBase + 8 ], z8);
    G     = wmma4(Fo[2], Fo[3], W[RP + bBase + 32], W[RP + bBase + 40], G);
    {   // store D-layout -> R1
      const int base = dval ? (R1 + eD * 64 + dcol) : (RT + (lane & 15));
      #pragma unroll
      for (int i = 0; i < 8; ++i) W[base + i * 8] = G[i];
    }
    // ---- P' = G*F^T + Q ----
    v8f Pp = wmma4(W[R1 + aBase + 0], W[R1 + aBase + 1], Fo[0], Fo[1], Qd);
    Pp     = wmma4(W[R1 + aBase + 4], W[R1 + aBase + 5], Fo[2], Fo[3], Pp);
    {
      const int base = dval ? (R2 + eD * 64 + dcol) : (RT + (lane & 15));
      #pragma unroll
      for (int i = 0; i < 8; ++i) W[base + i * 8] = Pp[i];
    }
    // ---- T1 = H*P' ----
    v8f T1 = wmma4(Ho[0], Ho[1], W[R2 + bBase + 0 ], W[R2 + bBase + 8 ], z8);
    T1     = wmma4(Ho[2], Ho[3], W[R2 + bBase + 32], W[R2 + bBase + 40], T1);
    {
      const int base = dval ? (R3 + eD * 32 + dcol) : (RT + (lane & 15));
      #pragma unroll
      for (int i = 0; i < 4; ++i) W[base + i * 8] = T1[i];
    }
    // ---- S = T1*H^T + R ;  PHt = P'*H^T ----
    v8f Sm = wmma4(W[R3 + a3Base + 0], W[R3 + a3Base + 1], Ho[0], Ho[1], Rd);
    Sm     = wmma4(W[R3 + a3Base + 4], W[R3 + a3Base + 5], Ho[2], Ho[3], Sm);
    v8f PHt = wmma4(W[R2 + aBase + 0], W[R2 + aBase + 1], Ho[0], Ho[1], z8);
    PHt     = wmma4(W[R2 + aBase + 4], W[R2 + aBase + 5], Ho[2], Ho[3], PHt);
    {
      const int base = (dval && dcol < 4) ? (R4 + eD * 32 + dcol) : (RT + (lane & 15));
      #pragma unroll
      for (int i = 0; i < 8; ++i) W[base + i * 4] = PHt[i];
    }

    // ---- K row solve: lane (24e + r) solves S y = PHt[r,:]  (S is SPD) ----
    const float4 rhs4 = *reinterpret_cast<const float4*>(&W[R4 + (eD << 5) + (r << 2)]);
    float rr4[4] = {rhs4.x, rhs4.y, rhs4.z, rhs4.w};
    float s[4][4];
    const int sS = 24 * eD;
    #pragma unroll
    for (int i = 0; i < 4; ++i)
      #pragma unroll
      for (int j = 0; j < 4; ++j) s[i][j] = shf(Sm[i], sS + j);
    #pragma unroll
    for (int k = 0; k < 4; ++k) {
      const float inv = __builtin_amdgcn_rcpf(s[k][k]);   // v_rcp_f32; S has R=0.1I floor
      #pragma unroll
      for (int j = k + 1; j < 4; ++j) s[k][j] *= inv;
      rr4[k] *= inv;
      #pragma unroll
      for (int i = k + 1; i < 4; ++i) {
        const float f = s[i][k];
        #pragma unroll
        for (int j = k + 1; j < 4; ++j) s[i][j] = fmaf(-f, s[k][j], s[i][j]);
        rr4[i] = fmaf(-f, rr4[k], rr4[i]);
      }
    }
    float Kp[4];
    Kp[3] = rr4[3];
    Kp[2] = fmaf(-s[2][3], Kp[3], rr4[2]);
    Kp[1] = fmaf(-s[1][3], Kp[3], fmaf(-s[1][2], Kp[2], rr4[1]));
    Kp[0] = fmaf(-s[0][3], Kp[3], fmaf(-s[0][2], Kp[2], fmaf(-s[0][1], Kp[1], rr4[0])));

    // ---- innovation + state update ----
    const long zidx = ((b0 + (lane >> 4)) * TT + t) * N4 + (r & 3);
    const float zv = z[zidx];
    __builtin_prefetch(&z[(t < TT - 8) ? (zidx + 8 * N4) : zidx], 0, 0);
    const float dy = zv - ys;                     // rep layout, valid for i<4
    const int sE = lane & 16;
    float xn = shf(xp, sE + r);
    #pragma unroll
    for (int j = 0; j < 4; ++j) xn = fmaf(Kp[j], shf(dy, sE + j), xn);
    if (dval) out[((b0 + eD) * TT + t) * 8 + r] = xn;

    // ---- P_new = P' - K*T1  (one WMMA, K contraction = 4, negate K in A) ----
    const int sK = 24 * e8 + r;                   // K-lane / T1 D-layout source lane
    float Av0, Av1, Bv0, Bv1;
    {
      const float klo0 = shf(Kp[0], sK), khi0 = shf(Kp[2], sK);
      const float klo1 = shf(Kp[1], sK), khi1 = shf(Kp[3], sK);
      Av0 = -(lowh ? klo0 : khi0);
      Av1 = -(lowh ? klo1 : khi1);
      const float tlo0 = shf(T1[0], sK), thi0 = shf(T1[2], sK);
      const float tlo1 = shf(T1[1], sK), thi1 = shf(T1[3], sK);
      Bv0 = lowh ? tlo0 : thi0;
      Bv1 = lowh ? tlo1 : thi1;
    }
    v8f Pn = wmma4(Av0, Av1, Bv0, Bv1, Pp);
    {
      const int base = dval ? (RP + eD * 64 + dcol) : (RT + (lane & 15));
      #pragma unroll
      for (int i = 0; i < 8; ++i) W[base + i * 8] = Pn[i];
    }
    xrep = shf(xn, 24 * (lane >> 4) + r);         // back to rep layout
  }
}

extern "C" void kernel_launch(void* const* d_in, const int* in_sizes, int n_in,
                              void* d_out, int out_size, void* d_ws, size_t ws_size,
                              hipStream_t stream) {
  const float* z  = (const float*)d_in[0];
  const float* A  = (const float*)d_in[1];
  const float* C  = (const float*)d_in[2];
  const float* Q  = (const float*)d_in[3];
  const float* R  = (const float*)d_in[4];
  const float* x0 = (const float*)d_in[5];
  const float* P0 = (const float*)d_in[6];
  float* out = (float*)d_out;

  const int Bsz = in_sizes[0] / (TT * N4);        // z is [B, T, 4]
  const int waves = Bsz / 2;                      // 2 chains per wave32
  const int threads = 256;
  int blocks = (waves * 32) / threads;
  if (blocks < 1) blocks = 1;
  ekf1250_kernel<<<blocks, threads, 0, stream>>>(z, A, C, Q, R, x0, P0, out);
}